// MetaLearner_61332132987355
// MI455X (gfx1250) — compile-verified
//
#include <hip/hip_runtime.h>
#include <math.h>

typedef __attribute__((ext_vector_type(2))) float v2f;
typedef __attribute__((ext_vector_type(8))) float v8f;

// ---------------------------------------------------------------------------
// Kernel 1: task embedding + 3 hierarchical soft-cluster layers (tiny).
// One block of 256 threads. Writes cated[128] = {task_emb, c3} into d_ws.
// ---------------------------------------------------------------------------
__global__ __launch_bounds__(256) void ml_small_kernel(
    const float* __restrict__ emb,        // [200000,32]
    const float* __restrict__ oh_x,       // [8]
    const float* __restrict__ mh_x,       // [2,50]
    const float* __restrict__ ctns,       // [8]
    const float* __restrict__ taskemb_W,  // [64,328]
    const float* __restrict__ centers1, const float* __restrict__ lin1_W, const float* __restrict__ lin1_b,
    const float* __restrict__ centers2, const float* __restrict__ lin2_W, const float* __restrict__ lin2_b,
    const float* __restrict__ centers3, const float* __restrict__ lin3_W, const float* __restrict__ lin3_b,
    const int* __restrict__ oh_i,         // [8]
    const int* __restrict__ mh_i,         // [2,50]
    float* __restrict__ cated_out)        // [128] in workspace
{
  __shared__ float sx[328];      // user feature vector
  __shared__ float te[64];       // task embedding
  __shared__ float cbuf[2][64];  // cluster ping-pong
  __shared__ float ss[8];        // softmax weights

  const int tid = threadIdx.x;

  // ---- stage A: build x[328] ----
  {
    int e = tid >> 5, d = tid & 31;                 // 256 threads -> 8x32 onehot
    sx[tid] = emb[(long long)oh_i[e] * 32 + d] * oh_x[e];
  }
  if (tid < 64) {                                   // 2x32 multihot (sum over 50)
    int r = tid >> 5, d = tid & 31;
    float acc = 0.0f;
    for (int j = 0; j < 50; ++j)
      acc += emb[(long long)mh_i[r * 50 + j] * 32 + d] * mh_x[r * 50 + j];
    sx[256 + tid] = acc;
  }
  if (tid < 8) sx[320 + tid] = ctns[tid];
  __syncthreads();

  // ---- stage B: task_emb = taskemb_W @ x ----
  if (tid < 64) {
    const float* wr = taskemb_W + tid * 328;
    float acc = 0.0f;
    for (int j = 0; j < 328; ++j) acc += wr[j] * sx[j];
    te[tid] = acc;
  }
  __syncthreads();

  // ---- stage C: 3 cluster layers (runtime-selected pointers, no LDS-ptr arrays)
  for (int L = 0; L < 3; ++L) {
    const int K = (L == 0) ? 8 : (L == 1) ? 4 : 1;
    const float* C = (L == 0) ? centers1 : (L == 1) ? centers2 : centers3;
    const float* W = (L == 0) ? lin1_W   : (L == 1) ? lin2_W   : lin3_W;
    const float* B = (L == 0) ? lin1_b   : (L == 1) ? lin2_b   : lin3_b;
    const float* cin = (L == 0) ? te : cbuf[(L + 1) & 1];
    float* cout = cbuf[L & 1];

    if (tid == 0) {  // softmax(-||centers - x||) ; K <= 8, trivial
      float dist[8], mn = 1e30f;
      for (int k = 0; k < K; ++k) {
        float d2 = 0.0f;
        for (int j = 0; j < 64; ++j) { float df = C[k * 64 + j] - cin[j]; d2 += df * df; }
        dist[k] = sqrtf(d2);
        mn = fminf(mn, dist[k]);
      }
      float sum = 0.0f;
      for (int k = 0; k < K; ++k) { float e = expf(mn - dist[k]); ss[k] = e; sum += e; }
      float inv = 1.0f / sum;
      for (int k = 0; k < K; ++k) ss[k] *= inv;
    }
    __syncthreads();

    if (tid < 64) {  // out[i] = sum_k s[k] * tanh(W[k,i,:] @ x + b[k,i])
      float o = 0.0f;
      for (int k = 0; k < K; ++k) {
        const float* wr = W + ((long long)k * 64 + tid) * 64;
        float acc = B[k * 64 + tid];
        for (int j = 0; j < 64; ++j) acc += wr[j] * cin[j];
        o += ss[k] * tanhf(acc);
      }
      cout[tid] = o;
    }
    __syncthreads();
  }

  // ---- emit cated = [task_emb, c] ; layer 3 output landed in cbuf[0] ----
  if (tid < 64)       cated_out[tid] = te[tid];
  else if (tid < 128) cated_out[tid] = cbuf[0][tid - 64];
}

// ---------------------------------------------------------------------------
// Kernel 2: out = sigmoid(adapt_W @ cated + b) * params   (HBM-bound GEMV)
// 128 threads = 4 waves; each wave handles 16 rows with 32x
// v_wmma_f32_16x16x4_f32 over K=128.  B operand = cated replicated across
// all 16 N-columns, so every column of D equals the GEMV result.
// A layout (ISA 7.12.2, 32-bit A 16x4): lanes 0-15 -> M=lane, K={k,k+1};
// lanes 16-31 -> M=lane-16, K={k+2,k+3}  => one global_load_b64 per lane/step.
// ---------------------------------------------------------------------------
__global__ __launch_bounds__(128) void ml_adapt_wmma_kernel(
    const float* __restrict__ W,       // [P,128]
    const float* __restrict__ bias,    // [P]
    const float* __restrict__ params,  // [P]
    const float* __restrict__ cated,   // [128] (workspace)
    float* __restrict__ out,           // [P]
    long long P)
{
  __shared__ float sc[128];
  __shared__ float sy[64];

  const int tid = threadIdx.x;
  sc[tid] = cated[tid];
  __syncthreads();

  const int wave = tid >> 5;
  const int lane = tid & 31;
  const long long blockBase = (long long)blockIdx.x * 64;
  long long rowBase = blockBase + (long long)wave * 16;
  if (rowBase + 16 > P) rowBase = P - 16;   // wave-uniform clamp (P%64==0 here)

  const int mrow  = lane & 15;
  const int khalf = (lane >> 4) << 1;       // 0 for lanes 0-15, 2 for 16-31
  const float* pA = W + (rowBase + mrow) * 128 + khalf;

  v8f acc = {};
#pragma unroll
  for (int t = 0; t < 32; ++t) {
    v2f a = *(const v2f*)(pA + 4 * t);      // global_load_b64 per lane
    v2f bm;
    bm.x = sc[4 * t + khalf];               // B[k][n] = cated[k] for all n
    bm.y = sc[4 * t + khalf + 1];
    acc = __builtin_amdgcn_wmma_f32_16x16x4_f32(
        /*neg_a=*/false, a, /*neg_b=*/false, bm,
        /*c_mod=*/(short)0, acc, /*reuse_a=*/false, /*reuse_b=*/false);
  }

  // Column N=0: lane 0 holds rows 0..7 in acc[0..7], lane 16 holds rows 8..15.
  if (lane == 0) {
#pragma unroll
    for (int i = 0; i < 8; ++i) sy[wave * 16 + i] = acc[i];
  } else if (lane == 16) {
#pragma unroll
    for (int i = 0; i < 8; ++i) sy[wave * 16 + 8 + i] = acc[i];
  }
  __syncthreads();

  if (tid < 64) {
    long long row = blockBase + tid;
    if (row < P) {
      float y = sy[tid] + bias[row];
      float g = 1.0f / (1.0f + __expf(-y));
      out[row] = g * params[row];
    }
  }
}

// ---------------------------------------------------------------------------
extern "C" void kernel_launch(void* const* d_in, const int* in_sizes, int n_in,
                              void* d_out, int out_size, void* d_ws, size_t ws_size,
                              hipStream_t stream) {
  const float* embeddings = (const float*)d_in[0];
  const float* onehot_x   = (const float*)d_in[1];
  const float* mh_x       = (const float*)d_in[2];
  const float* ctns       = (const float*)d_in[3];
  const float* taskemb_W  = (const float*)d_in[4];
  const float* centers1   = (const float*)d_in[5];
  const float* lin1_W     = (const float*)d_in[6];
  const float* lin1_b     = (const float*)d_in[7];
  const float* centers2   = (const float*)d_in[8];
  const float* lin2_W     = (const float*)d_in[9];
  const float* lin2_b     = (const float*)d_in[10];
  const float* centers3   = (const float*)d_in[11];
  const float* lin3_W     = (const float*)d_in[12];
  const float* lin3_b     = (const float*)d_in[13];
  const float* params_flat= (const float*)d_in[14];
  const float* adapt_W    = (const float*)d_in[15];
  const float* adapt_b    = (const float*)d_in[16];
  const int*   onehot_i   = (const int*)d_in[17];
  const int*   mh_i       = (const int*)d_in[18];

  float* out   = (float*)d_out;
  float* cated = (float*)d_ws;   // 128 floats of scratch

  ml_small_kernel<<<1, 256, 0, stream>>>(
      embeddings, onehot_x, mh_x, ctns, taskemb_W,
      centers1, lin1_W, lin1_b, centers2, lin2_W, lin2_b,
      centers3, lin3_W, lin3_b, onehot_i, mh_i, cated);

  const long long P = (long long)out_size;          // 1,000,000 (64 | P)
  const int grid = (int)((P + 63) / 64);            // 15625 blocks, 4 waves each
  ml_adapt_wmma_kernel<<<grid, 128, 0, stream>>>(
      adapt_W, adapt_b, params_flat, cated, out, P);
}